// Expansion_20839181320729
// MI455X (gfx1250) — compile-verified
//
#include <hip/hip_runtime.h>
#include <math.h>
#include <stdint.h>

typedef __attribute__((ext_vector_type(2))) float v2f;
typedef __attribute__((ext_vector_type(8))) float v8f;

// ---------------- problem constants ----------------
// IRREP_IN  = 128 x (l=0..4), DIM_IN  = 3200
// IRREP_OUT =  64 x (l=0..2), DIM_OUT = 576
// 19 instructions, each with 128*64*64 weights.

__host__ __device__ constexpr int off_in(int li) {
  return li == 0 ? 0 : li == 1 ? 128 : li == 2 ? 512 : li == 3 ? 1152 : 2048;
}
__host__ __device__ constexpr int blk_off(int l) { return l == 0 ? 0 : l == 1 ? 64 : 256; }

// Enumeration identical to the Python INSTRUCTIONS loop (i outer, then j, then k).
__host__ __device__ constexpr int instr_index(int li, int j, int k) {
  int t = 0;
  for (int i = 0; i < 5; ++i)
    for (int jj = 0; jj < 3; ++jj)
      for (int kk = 0; kk < 3; ++kk) {
        int lo = jj > kk ? jj - kk : kk - jj;
        if (lo <= i && i <= jj + kk) {
          if (i == li && jj == j && kk == k) return t;
          ++t;
        }
      }
  return -1;
}

// ---------------- CDNA5 async global->LDS support ----------------
#if defined(__AMDGCN__) && __has_builtin(__builtin_amdgcn_global_load_async_to_lds_b32)
#define HAS_ASYNC_LDS 1
#else
#define HAS_ASYNC_LDS 0
#endif

// Builtin params are addrspace-qualified int* (clang prints AS1 as "__device__",
// AS3 as "__shared__" in HIP diagnostics).
#define AS1P(p) ((__attribute__((address_space(1))) int*)(uintptr_t)(p))
#define AS3P(p) ((__attribute__((address_space(3))) int*)(uint32_t)(uintptr_t)(p))

__device__ __forceinline__ void wait_async0() {
#if HAS_ASYNC_LDS
#if __has_builtin(__builtin_amdgcn_s_wait_asynccnt)
  __builtin_amdgcn_s_wait_asynccnt(0);
#else
  asm volatile("s_wait_asynccnt 0" ::: "memory");
#endif
#endif
}

// =====================================================================
// Setup kernel: compute real-basis Wigner 3j tables (same convention as
// the reference) in double precision, one block per instruction.
// Layout in workspace: slot t = 256 floats, [i(d1)][jj(d2)][k3(d3)].
// =====================================================================
struct cplx { double re, im; };
__device__ inline cplx cmul(cplx a, cplx b) {
  return {a.re * b.re - a.im * b.im, a.re * b.im + a.im * b.re};
}
__device__ inline cplx conjc(cplx a) { return {a.re, -a.im}; }

__device__ double dfact(int n) { double r = 1.0; for (int i = 2; i <= n; ++i) r *= i; return r; }

__device__ double su2_cg(int j1, int m1, int j2, int m2, int j3, int m3) {
  if (m3 != m1 + m2) return 0.0;
  int vmin = -j1 + j2 + m3;
  if (-j1 + m1 > vmin) vmin = -j1 + m1;
  if (0 > vmin) vmin = 0;
  int vmax = j2 + j3 + m1;
  if (j3 - j1 + j2 < vmax) vmax = j3 - j1 + j2;
  if (j3 + m3 < vmax) vmax = j3 + m3;
  double c = sqrt((2.0 * j3 + 1.0) * dfact(j3 + j1 - j2) * dfact(j3 - j1 + j2) *
                  dfact(j1 + j2 - j3) * dfact(j3 + m3) * dfact(j3 - m3) /
                  (dfact(j1 + j2 + j3 + 1) * dfact(j1 - m1) * dfact(j1 + m1) *
                   dfact(j2 - m2) * dfact(j2 + m2)));
  double s = 0.0;
  for (int v = vmin; v <= vmax; ++v) {
    double sg = ((v + j2 + m2) & 1) ? -1.0 : 1.0;
    s += sg * dfact(j2 + j3 + m1 - v) * dfact(j1 - m1 + v) /
         (dfact(v) * dfact(j3 - j1 + j2 - v) * dfact(j3 + m3 - v) * dfact(v + j1 - j2 - m3));
  }
  return c * s;
}

__device__ void real_to_complex(int l, cplx q[9][9]) {
  for (int r = 0; r < 9; ++r)
    for (int c = 0; c < 9; ++c) q[r][c] = {0.0, 0.0};
  const double is2 = 0.70710678118654752440;
  for (int m = -l; m < 0; ++m) {
    q[l + m][l - m] = {is2, 0.0};    // q[l+m, l+|m|] = 1/sqrt2
    q[l + m][l + m] = {0.0, -is2};   // q[l+m, l-|m|] = -i/sqrt2
  }
  q[l][l] = {1.0, 0.0};
  for (int m = 1; m <= l; ++m) {
    double sg = (m & 1) ? -1.0 : 1.0;
    q[l + m][l + m] = {sg * is2, 0.0};
    q[l + m][l - m] = {0.0, sg * is2};
  }
  int lm = l & 3;  // multiply by (-i)^l
  cplx ph = (lm == 0) ? cplx{1.0, 0.0} : (lm == 1) ? cplx{0.0, -1.0}
          : (lm == 2) ? cplx{-1.0, 0.0} : cplx{0.0, 1.0};
  for (int r = 0; r < 9; ++r)
    for (int c = 0; c < 9; ++c) q[r][c] = cmul(q[r][c], ph);
}

__global__ void setup_w3j(float* __restrict__ ws) {
  const int t = blockIdx.x;  // 0..18
  int l1 = 0, l2 = 0, l3 = 0;
  {
    int c = 0;
    for (int i = 0; i < 5; ++i)
      for (int j = 0; j < 3; ++j)
        for (int k = 0; k < 3; ++k) {
          int lo = j > k ? j - k : k - j;
          if (lo <= i && i <= j + k) {
            if (c == t) { l3 = i; l1 = j; l2 = k; }
            ++c;
          }
        }
  }
  __shared__ cplx Q1[9][9], Q2[9][9], Q3[9][9];
  __shared__ double CG[5][5][9];
  if (threadIdx.x == 0) {
    real_to_complex(l1, Q1);
    real_to_complex(l2, Q2);
    real_to_complex(l3, Q3);
    for (int a = 0; a < 5; ++a)
      for (int b = 0; b < 5; ++b)
        for (int c = 0; c < 9; ++c) CG[a][b][c] = 0.0;
    double inv = 1.0 / sqrt(2.0 * l3 + 1.0);
    for (int m1 = -l1; m1 <= l1; ++m1)
      for (int m2 = -l2; m2 <= l2; ++m2) {
        int m3 = m1 + m2;
        if (m3 >= -l3 && m3 <= l3)
          CG[l1 + m1][l2 + m2][l3 + m3] = su2_cg(l1, m1, l2, m2, l3, m3) * inv;
      }
  }
  __syncthreads();
  const int d1 = 2 * l1 + 1, d2 = 2 * l2 + 1, d3 = 2 * l3 + 1, n = d1 * d2 * d3;
  for (int idx = threadIdx.x; idx < 256; idx += blockDim.x) {
    float val = 0.0f;
    if (idx < n) {
      int io = idx / (d2 * d3);
      int rem = idx % (d2 * d3);
      int lo_ = rem / d3;
      int mo = rem % d3;
      double sre = 0.0;
      // W[jlm] = Re( sum_{i,k,n} Q1[i,j] Q2[k,l] conj(Q3[n,m]) CG[i,k,n] )
      for (int i = 0; i < d1; ++i)
        for (int k = 0; k < d2; ++k) {
          cplx q12 = cmul(Q1[i][io], Q2[k][lo_]);
          for (int nn = 0; nn < d3; ++nn) {
            double cg = CG[i][k][nn];
            if (cg != 0.0) {
              cplx term = cmul(q12, conjc(Q3[nn][mo]));
              sre += term.re * cg;
            }
          }
        }
      val = (float)sre;
    }
    ws[t * 256 + idx] = val;
  }
}

// =====================================================================
// Main kernel: one template instantiation per output block (j,k).
// Workgroup = (u, b-tile of 16); 4 waves take v-tiles 0/16/32/48 and
// share double-buffered X panels in LDS (async global->LDS copies
// overlapped with WMMA compute).  GEMM via V_WMMA_F32_16X16X4_F32:
//   D[v_loc, b_loc] += sum_w W[w, u, v0+v_loc] * X[b0+b_loc, w, k3]
// WMMA layout assumptions (f32):
//   A per lane: M = lane&15, K = 2*(lane>>4) + component   (ISA table)
//   B per lane: K = 2*(lane>>4) + component, N = lane&15   (symmetric)
//   D per lane: reg r -> M = r + 8*(lane>>4), N = lane&15  (ISA table)
// =====================================================================

#define XPANEL (16 * 132)  // padded stride 132 -> conflict-free ds_load_b64

// Issue one X panel (16 b-rows x 128 w) for slice k3 into sXbuf.
// Thread tid owns column w = tid; rows bb = 0..15 (uniform 12800B steps).
template <int DK, int XOFF>
__device__ __forceinline__ void issue_panel(const float* __restrict__ x,
                                            float* sXbuf, int b0, int k3) {
  const int tid = threadIdx.x;
  const float* gp = x + b0 * 3200 + XOFF + tid * DK + k3;
#if HAS_ASYNC_LDS
  float* lp = sXbuf + tid;
#pragma unroll
  for (int it = 0; it < 16; ++it) {
    __builtin_amdgcn_global_load_async_to_lds_b32(AS1P(gp + it * 3200),
                                                  AS3P(lp + it * 132), 0, 0);
  }
#else
#pragma unroll
  for (int it = 0; it < 16; ++it) sXbuf[it * 132 + tid] = gp[it * 3200];
#endif
}

template <int LJ, int LK, int LI>
__device__ __forceinline__ void process_li(const float* __restrict__ x,
                                           const float* __restrict__ w,
                                           float* sX,  // 2 * XPANEL floats
                                           const float* sW3J_all,
                                           int u, int b0, int v0, int l15, int h,
                                           v8f (&acc)[(2 * LJ + 1) * (2 * LK + 1)]) {
  constexpr int LMIN = (LJ > LK) ? LJ - LK : LK - LJ;
  constexpr int LMAX = LJ + LK;
  if constexpr (LI >= LMIN && LI <= LMAX) {
    constexpr int D1 = 2 * LJ + 1, D2 = 2 * LK + 1, DK = 2 * LI + 1;
    constexpr int T = instr_index(LI, LJ, LK);
    constexpr int XOFF = off_in(LI);
    const float* sW = sW3J_all + (LI - LMIN) * 256;

    // A fragments for the whole K=128 loop (independent of k3).
    const float* wbase = w + (size_t)T * 524288 + u * 64 + v0 + l15;
    const int krow = 2 * h;
    v2f a[32];
#pragma unroll
    for (int i = 0; i < 32; ++i) {
      a[i].x = wbase[(size_t)(4 * i + krow) * 4096];
      a[i].y = wbase[(size_t)(4 * i + krow + 1) * 4096];
    }

    __syncthreads();  // all waves done reading both buffers (prev li / sW3J fence)
    issue_panel<DK, XOFF>(x, sX, b0, 0);

    for (int k3 = 0; k3 < DK; ++k3) {
      float* cur = sX + (k3 & 1) * XPANEL;
      wait_async0();    // this wave's panel-k3 copies landed in LDS
      __syncthreads();  // everyone's landed; everyone done reading other buffer
      if (k3 + 1 < DK)  // prefetch next slice under the WMMA work below
        issue_panel<DK, XOFF>(x, sX + ((k3 + 1) & 1) * XPANEL, b0, k3 + 1);

      v8f d = {0.f, 0.f, 0.f, 0.f, 0.f, 0.f, 0.f, 0.f};
#pragma unroll
      for (int w0 = 0; w0 < 32; ++w0) {
        v2f bf;
        bf.x = cur[l15 * 132 + 4 * w0 + krow];
        bf.y = cur[l15 * 132 + 4 * w0 + krow + 1];
        d = __builtin_amdgcn_wmma_f32_16x16x4_f32(false, a[w0], false, bf,
                                                  (short)0, d, false, false);
      }
      // Fold tmp[.,.,k3] into output accumulators with Wigner-3j weights.
#pragma unroll
      for (int i = 0; i < D1; ++i)
#pragma unroll
        for (int jj = 0; jj < D2; ++jj) {
          float c3 = sW[(i * D2 + jj) * DK + k3];
          acc[i * D2 + jj] += d * c3;
        }
    }
  }
}

template <int LJ, int LK>
__global__ __launch_bounds__(128, 1) void expansion_block(
    const float* __restrict__ x, const float* __restrict__ w,
    const float* __restrict__ w3j, float* __restrict__ out) {
  constexpr int D1 = 2 * LJ + 1, D2 = 2 * LK + 1;
  constexpr int LMIN = (LJ > LK) ? LJ - LK : LK - LJ;
  constexpr int LMAX = LJ + LK;
  constexpr int NLI = LMAX - LMIN + 1;

  __shared__ float sX[2 * XPANEL];
  __shared__ float sW3J[NLI * 256];

  const int u = blockIdx.x >> 4;           // 0..63
  const int b0 = (blockIdx.x & 15) << 4;   // batch tile base
  const int lane = threadIdx.x & 31;
  const int wave = threadIdx.x >> 5;
  const int v0 = wave << 4;                // v tile per wave
  const int l15 = lane & 15;
  const int h = lane >> 4;

  for (int idx = threadIdx.x; idx < NLI * 256; idx += 128) {
    int li = LMIN + (idx >> 8);
    sW3J[idx] = w3j[instr_index(li, LJ, LK) * 256 + (idx & 255)];
  }

  v8f acc[D1 * D2];
  const v8f vz = {0.f, 0.f, 0.f, 0.f, 0.f, 0.f, 0.f, 0.f};
#pragma unroll
  for (int t = 0; t < D1 * D2; ++t) acc[t] = vz;

  // (the barrier at the head of each process_li fences the sW3J preload)
  process_li<LJ, LK, 0>(x, w, sX, sW3J, u, b0, v0, l15, h, acc);
  process_li<LJ, LK, 1>(x, w, sX, sW3J, u, b0, v0, l15, h, acc);
  process_li<LJ, LK, 2>(x, w, sX, sW3J, u, b0, v0, l15, h, acc);
  process_li<LJ, LK, 3>(x, w, sX, sW3J, u, b0, v0, l15, h, acc);
  process_li<LJ, LK, 4>(x, w, sX, sW3J, u, b0, v0, l15, h, acc);

  const float invm = 1.0f / 128.0f;
  float* ob = out + (size_t)(b0 + l15) * 576 * 576;
#pragma unroll
  for (int i = 0; i < D1; ++i) {
    const int row = blk_off(LJ) + u * D1 + i;
#pragma unroll
    for (int jj = 0; jj < D2; ++jj) {
      v8f aa = acc[i * D2 + jj];
#pragma unroll
      for (int r = 0; r < 8; ++r) {
        int v = v0 + r + 8 * h;
        ob[row * 576 + blk_off(LK) + v * D2 + jj] = aa[r] * invm;
      }
    }
  }
}

// =====================================================================
extern "C" void kernel_launch(void* const* d_in, const int* in_sizes, int n_in,
                              void* d_out, int out_size, void* d_ws, size_t ws_size,
                              hipStream_t stream) {
  (void)in_sizes; (void)n_in; (void)out_size; (void)ws_size;
  const float* x = (const float*)d_in[0];
  const float* w = (const float*)d_in[1];
  float* out = (float*)d_out;
  float* wsf = (float*)d_ws;  // 19*256 floats of Wigner-3j tables

  setup_w3j<<<19, 256, 0, stream>>>(wsf);

  dim3 grid(64 * 16), block(128);
  expansion_block<0, 0><<<grid, block, 0, stream>>>(x, w, wsf, out);
  expansion_block<0, 1><<<grid, block, 0, stream>>>(x, w, wsf, out);
  expansion_block<0, 2><<<grid, block, 0, stream>>>(x, w, wsf, out);
  expansion_block<1, 0><<<grid, block, 0, stream>>>(x, w, wsf, out);
  expansion_block<1, 1><<<grid, block, 0, stream>>>(x, w, wsf, out);
  expansion_block<1, 2><<<grid, block, 0, stream>>>(x, w, wsf, out);
  expansion_block<2, 0><<<grid, block, 0, stream>>>(x, w, wsf, out);
  expansion_block<2, 1><<<grid, block, 0, stream>>>(x, w, wsf, out);
  expansion_block<2, 2><<<grid, block, 0, stream>>>(x, w, wsf, out);
}